// TestModel_63848983823017
// MI455X (gfx1250) — compile-verified
//
#include <hip/hip_runtime.h>

// ---------------------------------------------------------------------------
// MI455X (gfx1250) implementation.
//
// Roofline: output state = 256*4*4*4096 complex64 = 134.2 MB of pure stores
// -> 23.3 TB/s floor ~= 5.8 us; everything else is negligible. The state
// factorizes into (complex qubit-0 2-vector, indexed by rank & MSB) x
// (real 11-qubit product p[2048] in LDS), so the kernel is a streaming
// non-temporal b128 writer (write-once data; keep it out of the 192MB L2).
// The 4x4 Gram matrix M = <state| Z..Z |state> is computed honestly with
// V_WMMA_F32_16X16X4_F32 (K=4096, 8 waves x 128 WMMA per workgroup). Each
// wave's K-range has a constant state MSB, so the qubit-0 coefficient and
// the (-1)^msb part of z are hoisted; the low-bit parity of z is baked into
// a second LDS table zp[] = +-p[]. Inner loop: 2x ds_load_b64 + 2x pk_mul
// + 1x wmma.
// ---------------------------------------------------------------------------

typedef float v2f __attribute__((ext_vector_type(2)));
typedef float v4f __attribute__((ext_vector_type(4)));
typedef float v8f __attribute__((ext_vector_type(8)));

#define NQ    12
#define DDIM  4096
#define LOWD  2048
#define BATCH 256
#define LBLK  4

struct cplx { float re, im; };
__device__ __forceinline__ cplx cmul(cplx a, cplx b) {
  return { a.re * b.re - a.im * b.im, a.re * b.im + a.im * b.re };
}

// Apply the k-th Schmidt factor of the cross-block CNOT on qubit 0.
// l==0: RZ_{P/M} = diag(w, conj(w)), w = exp(-i*pi/4) (k=0) / exp(+i*pi/4) (k=1)
// l==1: RX_{P/M} = (1/sqrt2)[[1, (-+)i],[(-+)i, 1]]  (k=0 -> -i, k=1 -> +i)
__device__ __forceinline__ void apply_gate(int l, int k, cplx& a, cplx& b) {
  const float R2 = 0.70710678118654752f;
  if (l == 0) {
    cplx w  = k ? cplx{R2,  R2} : cplx{R2, -R2};
    cplx wc = k ? cplx{R2, -R2} : cplx{R2,  R2};
    a = cmul(a, w);
    b = cmul(b, wc);
  } else {
    float s = k ? 1.f : -1.f;                 // sign of i in the off-diagonal
    cplx ib = { -s * b.im, s * b.re };        // (s*i)*b
    cplx ia = { -s * a.im, s * a.re };        // (s*i)*a
    cplx na = { R2 * (a.re + ib.re), R2 * (a.im + ib.im) };
    cplx nb = { R2 * (b.re + ia.re), R2 * (b.im + ia.im) };
    a = na; b = nb;
  }
}

__global__ __launch_bounds__(256) void qstate_kernel(const float* __restrict__ x,
                                                     float* __restrict__ out_state,
                                                     float2* __restrict__ Mws)
{
  __shared__ __align__(16) float p[LOWD];    // 11-qubit real product table
  __shared__ __align__(16) float zp[LOWD];   // z-low-parity-signed copy of p
  __shared__ float cv[2][4][2];              // [Re/Im][rank][msb] qubit-0 coeffs
  __shared__ float red[8 * 256];             // cross-wave WMMA reduction
  __shared__ float G8[8][8];                 // real Gram G = (z.X) X^T

  const int l   = blockIdx.x;
  const int b   = blockIdx.y;
  const int tid = threadIdx.x;

  const float* x0 = x + (size_t)(b * LBLK + l) * NQ;          // x[0,b,l,:]
  const float* x1 = x0 + (size_t)BATCH * LBLK * NQ;           // x[1,b,l,:]

  // Combined half-angles for qubits 1..11 (Ry(t0)Ry(t1) = Ry(t0+t1)).
  float cq[NQ], sq[NQ];
#pragma unroll
  for (int q = 1; q < NQ; ++q) {
    float h = 0.5f * (x0[q] + x1[q]);
    cq[q] = __cosf(h);
    sq[q] = __sinf(h);
  }

  // p[m] = prod_{q=1..11} (bit_{11-q}(m) ? sin : cos); zp[m] = (-1)^popc(m) p[m]
  for (int m = tid; m < LOWD; m += 256) {
    float v = 1.f;
#pragma unroll
    for (int q = 1; q < NQ; ++q)
      v *= ((m >> (NQ - 1 - q)) & 1) ? sq[q] : cq[q];
    p[m]  = v;
    zp[m] = (__popc(m) & 1) ? -v : v;
  }

  if (tid == 0) {
    float h0 = 0.5f * x0[0], h1 = 0.5f * x1[0];
    if (l >= 2) {
      float ct = __cosf(h0 + h1), st = __sinf(h0 + h1);
#pragma unroll
      for (int r = 0; r < 4; ++r) {
        cv[0][r][0] = ct;  cv[0][r][1] = st;
        cv[1][r][0] = 0.f; cv[1][r][1] = 0.f;
      }
    } else {
      float c0 = __cosf(h0), s0 = __sinf(h0);
      float c1 = __cosf(h1), s1 = __sinf(h1);
#pragma unroll
      for (int k1 = 0; k1 < 2; ++k1)
#pragma unroll
        for (int k2 = 0; k2 < 2; ++k2) {
          cplx a = {c0, 0.f}, bv = {s0, 0.f};
          apply_gate(l, k1, a, bv);                 // first entangle factor
          cplx a2 = { c1 * a.re - s1 * bv.re, c1 * a.im - s1 * bv.im };
          cplx b2 = { s1 * a.re + c1 * bv.re, s1 * a.im + c1 * bv.im };
          apply_gate(l, k2, a2, b2);                // second entangle factor
          int r = 2 * k1 + k2;
          cv[0][r][0] = a2.re; cv[1][r][0] = a2.im;
          cv[0][r][1] = b2.re; cv[1][r][1] = b2.im;
        }
    }
  }
  __syncthreads();

  // ---- Stream the state: st[b,l,r,d] = cv[r][msb(d)] * p[d & 2047] ----
  // Non-temporal b128 stores: 134 MB write-once, keep it out of L2.
  const float2* p2 = (const float2*)p;
  v4f* o4 = (v4f*)out_state;
  const size_t cbase = (size_t)(b * LBLK + l) * 4 * DDIM;   // complex index
#pragma unroll
  for (int r = 0; r < 4; ++r) {
    v4f* dst = o4 + ((cbase + (size_t)r * DDIM) >> 1);      // v4f index
    float cre0 = cv[0][r][0], cim0 = cv[1][r][0];
    float cre1 = cv[0][r][1], cim1 = cv[1][r][1];
    for (int i = tid; i < 1024; i += 256) {                 // msb = 0 half
      float2 pp = p2[i];
      v4f v = { cre0 * pp.x, cim0 * pp.x, cre0 * pp.y, cim0 * pp.y };
      __builtin_nontemporal_store(v, dst + i);
    }
    for (int i = tid; i < 1024; i += 256) {                 // msb = 1 half
      float2 pp = p2[i];
      v4f v = { cre1 * pp.x, cim1 * pp.x, cre1 * pp.y, cim1 * pp.y };
      __builtin_nontemporal_store(v, dst + 1024 + i);
    }
  }

  // ---- Gram matrix via V_WMMA_F32_16X16X4_F32 ----
  // X rows 0..3 = Re(st_r), 4..7 = Im(st_r), rows 8..15 = 0. A = z (.) X
  // chunk (16x4), B = chunk^T (4x16). G = sum_K A B over K = 4096.
  // Wave w owns K in [w*512, (w+1)*512) -> state MSB = (w >= 4) is constant,
  // so hoist the row coefficient and the (-1)^msb part of z out of the loop.
  const int wave = tid >> 5, lane = tid & 31;
  const int n = lane & 15, half = lane >> 4;
  float crow0 = 0.f, crow1 = 0.f;
  if (n < 8) {
    int r = n & 3, part = n >> 2;                 // part: 0 = Re, 1 = Im
    crow0 = cv[part][r][0];
    crow1 = cv[part][r][1];
  }
  const bool  hi = (wave >= 4);
  const float cB = hi ?  crow1 : crow0;           // B-matrix row coefficient
  const float cA = hi ? -crow1 : crow0;           // includes (-1)^msb of z

  const float2* zp2 = (const float2*)zp;
  v8f acc = {0.f, 0.f, 0.f, 0.f, 0.f, 0.f, 0.f, 0.f};
  // float2 index into the low-2048 tables; advances by 2 elements (4 K) / iter
  int j = (((wave * 512) & (LOWD - 1)) + 2 * half) >> 1;
#pragma unroll 4
  for (int it = 0; it < 128; ++it, j += 2) {
    float2 pp = p2[j];                            // ds_load_b64
    float2 zz = zp2[j];                           // ds_load_b64
    v2f A = { cA * zz.x, cA * zz.y };             // v_pk_mul_f32
    v2f B = { cB * pp.x, cB * pp.y };             // v_pk_mul_f32
    acc = __builtin_amdgcn_wmma_f32_16x16x4_f32(
        /*neg_a=*/false, A, /*neg_b=*/false, B,
        /*c_mod=*/(short)0, acc, /*reuse_a=*/false, /*reuse_b=*/false);
  }

  // Reduce 8 per-wave C fragments in LDS.
#pragma unroll
  for (int g = 0; g < 8; ++g) red[wave * 256 + lane * 8 + g] = acc[g];
  __syncthreads();
  float s = 0.f;
#pragma unroll
  for (int w = 0; w < 8; ++w) s += red[w * 256 + tid];
  // thread t<64 holds G[m = t&7][n = t>>3] (C layout: VGPR g = row, lane = col)
  if (tid < 64) G8[tid & 7][tid >> 3] = s;
  __syncthreads();

  if (tid < 16) {
    int r = tid >> 2, sidx = tid & 3;
    float re = G8[r][sidx]     + G8[r + 4][sidx + 4];
    float im = G8[r][sidx + 4] - G8[r + 4][sidx];
    Mws[(size_t)(b * LBLK + l) * 16 + tid] = make_float2(re, im);
  }
}

// O[b] = sum_{r,s} conj(alpha_r) * (prod_l M[b,l,r,s]) * alpha_s
__global__ __launch_bounds__(256) void qobs_kernel(const float2* __restrict__ Mws,
                                                   float* __restrict__ outO)
{
  int b = threadIdx.x;   // single 256-thread block
  // alpha = kron(A0, A0), A0 = [e^{i pi/4}, e^{-i pi/4}]/sqrt2
  const float2 alpha[4] = { {0.f, 0.5f}, {0.5f, 0.f}, {0.5f, 0.f}, {0.f, -0.5f} };
  float accr = 0.f, acci = 0.f;
#pragma unroll
  for (int r = 0; r < 4; ++r) {
#pragma unroll
    for (int si = 0; si < 4; ++si) {
      float pr = 1.f, pi = 0.f;                 // elementwise prod over blocks
#pragma unroll
      for (int l = 0; l < 4; ++l) {
        float2 m = Mws[(size_t)(b * LBLK + l) * 16 + r * 4 + si];
        float nr = pr * m.x - pi * m.y;
        float ni = pr * m.y + pi * m.x;
        pr = nr; pi = ni;
      }
      float ar = alpha[r].x, ai = -alpha[r].y;  // conj(alpha_r)
      float wr = ar * pr - ai * pi;
      float wi = ar * pi + ai * pr;
      accr += wr * alpha[si].x - wi * alpha[si].y;
      acci += wr * alpha[si].y + wi * alpha[si].x;
    }
  }
  outO[2 * b]     = accr;
  outO[2 * b + 1] = acci;
}

extern "C" void kernel_launch(void* const* d_in, const int* in_sizes, int n_in,
                              void* d_out, int out_size, void* d_ws, size_t ws_size,
                              hipStream_t stream) {
  (void)in_sizes; (void)n_in; (void)out_size; (void)ws_size;
  const float* x = (const float*)d_in[0];
  float*  out = (float*)d_out;
  float2* Mws = (float2*)d_ws;   // B*L*16 complex = 128 KB scratch

  dim3 grid(LBLK, BATCH);        // one workgroup per (block l, batch b)
  qstate_kernel<<<grid, 256, 0, stream>>>(x, out, Mws);

  const size_t stateFloats = (size_t)BATCH * LBLK * 4 * DDIM * 2; // 33,554,432
  qobs_kernel<<<1, 256, 0, stream>>>(Mws, out + stateFloats);
}